// MyModel_37434934952019
// MI455X (gfx1250) — compile-verified
//
#include <hip/hip_runtime.h>

#define N_NODE 50000
#define SEQN   32
#define NFEAT  4
#define HID    64
#define HEADS  8
#define OUTC   8

static_assert(N_NODE % 16 == 0, "tile math assumes N_NODE multiple of 16");

typedef float v2f __attribute__((ext_vector_type(2)));
typedef float v8f __attribute__((ext_vector_type(8)));

#define HAS_WMMA4 __has_builtin(__builtin_amdgcn_wmma_f32_16x16x4_f32)

__device__ __forceinline__ void atomicMaxF(float* addr, float v) {
  // Monotonic-bit-pattern trick: valid for arbitrary mixed-sign floats.
  if (v >= 0.0f) atomicMax((int*)addr, __float_as_int(v));
  else           atomicMin((unsigned int*)addr, __float_as_uint(v));
}

__device__ __forceinline__ float lrelu(float v) { return v > 0.0f ? v : 0.2f * v; }

// ---------------- per-step init of accumulators ----------------
__global__ void k_init(float* __restrict__ m1, float* __restrict__ den1,
                       float* __restrict__ agg1, float* __restrict__ m2,
                       float* __restrict__ den2, float* __restrict__ agg2) {
  int i = blockIdx.x * blockDim.x + threadIdx.x;
  if (i < N_NODE * HID)   agg1[i] = 0.0f;
  if (i < N_NODE * HEADS) { m1[i] = -3.0e38f; den1[i] = 0.0f; }
  if (i < N_NODE * NFEAT) agg2[i] = 0.0f;
  if (i < N_NODE)         { m2[i] = -3.0e38f; den2[i] = 0.0f; }
}

// ---------------- layer1 transform: xp1 = x_t @ W1 via WMMA 16x16x4 ----------------
// N_NODE % 16 == 0 -> every tile is full, no bounds checks needed.
__global__ void k_xform1(const float* __restrict__ xt, const float* __restrict__ W1,
                         float* __restrict__ xp1) {
  int wave = threadIdx.x >> 5;
  int lane = threadIdx.x & 31;
  int tile = blockIdx.x * (blockDim.x >> 5) + wave;
  int base = tile * 16;
  if (base >= N_NODE) return;                  // wave-uniform exit
  int row = lane & 15;                         // M index (A layout)
  int kh  = lane >> 4;                         // K-half select
  int col = lane & 15;                         // N index (B/D layout)
  int node = base + row;
#if HAS_WMMA4
  v2f a = *(const v2f*)(xt + node * NFEAT + 2 * kh);
  #pragma unroll
  for (int chunk = 0; chunk < 4; ++chunk) {
    v2f b;
    b.x = W1[(2 * kh + 0) * HID + chunk * 16 + col];
    b.y = W1[(2 * kh + 1) * HID + chunk * 16 + col];
    v8f c = {0.f, 0.f, 0.f, 0.f, 0.f, 0.f, 0.f, 0.f};
    c = __builtin_amdgcn_wmma_f32_16x16x4_f32(false, a, false, b, (short)0, c,
                                              false, false);
    float* outp = xp1 + (base + 8 * kh) * HID + chunk * 16 + col;
    #pragma unroll
    for (int j = 0; j < 8; ++j) outp[j * HID] = c[j];
  }
#else
  // Scalar fallback with identical wave/tile mapping.
  #pragma unroll
  for (int chunk = 0; chunk < 4; ++chunk) {
    #pragma unroll
    for (int j = 0; j < 8; ++j) {
      int n2 = base + j + 8 * kh;
      int cc = chunk * 16 + col;
      float s = 0.0f;
      for (int k = 0; k < NFEAT; ++k) s += xt[n2 * NFEAT + k] * W1[k * HID + cc];
      xp1[n2 * HID + cc] = s;
    }
  }
#endif
  (void)node;
}

// ---------------- attention logits, layer 1 ----------------
__global__ void k_alpha1(const float* __restrict__ xp1, const float* __restrict__ as,
                         const float* __restrict__ ad, float* __restrict__ asrc,
                         float* __restrict__ adst) {
  int i = blockIdx.x * blockDim.x + threadIdx.x;  // n*8 + h
  if (i >= N_NODE * HEADS) return;
  int n = i >> 3, h = i & 7;
  const float* xr = xp1 + n * HID + h * OUTC;
  float s1 = 0.0f, s2 = 0.0f;
  #pragma unroll
  for (int c = 0; c < OUTC; ++c) {
    float v = xr[c];
    s1 += v * as[h * OUTC + c];
    s2 += v * ad[h * OUTC + c];
  }
  asrc[i] = s1;
  adst[i] = s2;
}

// ---------------- edge pass A: per-dst max ----------------
__global__ void k_emax1(const int* __restrict__ src, const int* __restrict__ dst, int E,
                        const float* __restrict__ asrc, const float* __restrict__ adst,
                        float* __restrict__ m1) {
  int e = blockIdx.x * blockDim.x + threadIdx.x;
  int EE = E + N_NODE;
  if (e >= EE) return;
  int s, d;
  if (e < E) { s = src[e]; d = dst[e]; } else { s = e - E; d = s; }
  const float4* a0 = (const float4*)(asrc + s * 8);
  const float4* b0 = (const float4*)(adst + d * 8);
  float4 sa = a0[0], sb = a0[1], da = b0[0], db = b0[1];
  float ev[8] = {sa.x + da.x, sa.y + da.y, sa.z + da.z, sa.w + da.w,
                 sb.x + db.x, sb.y + db.y, sb.z + db.z, sb.w + db.w};
  #pragma unroll
  for (int h = 0; h < 8; ++h) atomicMaxF(&m1[d * 8 + h], lrelu(ev[h]));
}

// ---------------- edge pass B: softmax denominator ----------------
__global__ void k_eden1(const int* __restrict__ src, const int* __restrict__ dst, int E,
                        const float* __restrict__ asrc, const float* __restrict__ adst,
                        const float* __restrict__ m1, float* __restrict__ den1) {
  int e = blockIdx.x * blockDim.x + threadIdx.x;
  int EE = E + N_NODE;
  if (e >= EE) return;
  int s, d;
  if (e < E) { s = src[e]; d = dst[e]; } else { s = e - E; d = s; }
  const float4* a0 = (const float4*)(asrc + s * 8);
  const float4* b0 = (const float4*)(adst + d * 8);
  float4 sa = a0[0], sb = a0[1], da = b0[0], db = b0[1];
  float ev[8] = {sa.x + da.x, sa.y + da.y, sa.z + da.z, sa.w + da.w,
                 sb.x + db.x, sb.y + db.y, sb.z + db.z, sb.w + db.w};
  #pragma unroll
  for (int h = 0; h < 8; ++h)
    atomicAdd(&den1[d * 8 + h], __expf(lrelu(ev[h]) - m1[d * 8 + h]));
}

// ---------------- edge pass C: weighted message scatter (thread per edge,head) ----------
__global__ void k_escat1(const int* __restrict__ src, const int* __restrict__ dst, int E,
                         const float* __restrict__ asrc, const float* __restrict__ adst,
                         const float* __restrict__ m1, const float* __restrict__ den1,
                         const float* __restrict__ xp1, float* __restrict__ agg1) {
  int i = blockIdx.x * blockDim.x + threadIdx.x;  // edge*8 + h
  int EE8 = (E + N_NODE) * 8;
  if (i >= EE8) return;
  int e = i >> 3, h = i & 7;
  int s, d;
  if (e < E) { s = src[e]; d = dst[e]; } else { s = e - E; d = s; }
  float v = lrelu(asrc[s * 8 + h] + adst[d * 8 + h]);
  float alpha = __expf(v - m1[d * 8 + h]) / (den1[d * 8 + h] + 1e-16f);
  const float4* xr = (const float4*)(xp1 + s * HID + h * OUTC);
  float4 x0 = xr[0], x1 = xr[1];
  float* og = agg1 + d * HID + h * OUTC;
  atomicAdd(&og[0], alpha * x0.x); atomicAdd(&og[1], alpha * x0.y);
  atomicAdd(&og[2], alpha * x0.z); atomicAdd(&og[3], alpha * x0.w);
  atomicAdd(&og[4], alpha * x1.x); atomicAdd(&og[5], alpha * x1.y);
  atomicAdd(&og[6], alpha * x1.z); atomicAdd(&og[7], alpha * x1.w);
}

// ---------------- bias + ELU ----------------
__global__ void k_elu(const float* __restrict__ agg1, const float* __restrict__ bias1,
                      float* __restrict__ h1) {
  int i = blockIdx.x * blockDim.x + threadIdx.x;
  if (i >= N_NODE * HID) return;
  float v = agg1[i] + bias1[i & 63];
  h1[i] = v > 0.0f ? v : __expf(v) - 1.0f;
}

// ---------------- layer2 transform: y2p = h1 @ W2 (64x4, zero-padded to 64x16) ----------
__global__ void k_xform2(const float* __restrict__ h1, const float* __restrict__ W2,
                         float* __restrict__ y2p) {
  int wave = threadIdx.x >> 5;
  int lane = threadIdx.x & 31;
  int tile = blockIdx.x * (blockDim.x >> 5) + wave;
  int base = tile * 16;
  if (base >= N_NODE) return;
  int row = lane & 15;
  int kh  = lane >> 4;
  int col = lane & 15;
  int node = base + row;
#if HAS_WMMA4
  v8f c = {0.f, 0.f, 0.f, 0.f, 0.f, 0.f, 0.f, 0.f};
  #pragma unroll
  for (int k4 = 0; k4 < 16; ++k4) {
    int k0 = k4 * 4 + 2 * kh;
    v2f a = *(const v2f*)(h1 + node * HID + k0);
    v2f b;
    b.x = (col < NFEAT) ? W2[(k0 + 0) * NFEAT + col] : 0.0f;
    b.y = (col < NFEAT) ? W2[(k0 + 1) * NFEAT + col] : 0.0f;
    c = __builtin_amdgcn_wmma_f32_16x16x4_f32(false, a, false, b, (short)0, c,
                                              false, false);
  }
  if (col < NFEAT) {
    float* outp = y2p + (base + 8 * kh) * NFEAT + col;
    #pragma unroll
    for (int j = 0; j < 8; ++j) outp[j * NFEAT] = c[j];
  }
#else
  if (col < NFEAT) {
    #pragma unroll
    for (int j = 0; j < 8; ++j) {
      int n2 = base + j + 8 * kh;
      float s = 0.0f;
      for (int k = 0; k < HID; ++k) s += h1[n2 * HID + k] * W2[k * NFEAT + col];
      y2p[n2 * NFEAT + col] = s;
    }
  }
#endif
  (void)node;
}

// ---------------- attention logits, layer 2 ----------------
__global__ void k_alpha2(const float* __restrict__ y2p, const float* __restrict__ as,
                         const float* __restrict__ ad, float* __restrict__ asrc,
                         float* __restrict__ adst) {
  int n = blockIdx.x * blockDim.x + threadIdx.x;
  if (n >= N_NODE) return;
  float4 y = ((const float4*)(y2p + n * NFEAT))[0];
  asrc[n] = y.x * as[0] + y.y * as[1] + y.z * as[2] + y.w * as[3];
  adst[n] = y.x * ad[0] + y.y * ad[1] + y.z * ad[2] + y.w * ad[3];
}

__global__ void k_emax2(const int* __restrict__ src, const int* __restrict__ dst, int E,
                        const float* __restrict__ asrc, const float* __restrict__ adst,
                        float* __restrict__ m2) {
  int e = blockIdx.x * blockDim.x + threadIdx.x;
  int EE = E + N_NODE;
  if (e >= EE) return;
  int s, d;
  if (e < E) { s = src[e]; d = dst[e]; } else { s = e - E; d = s; }
  atomicMaxF(&m2[d], lrelu(asrc[s] + adst[d]));
}

__global__ void k_eden2(const int* __restrict__ src, const int* __restrict__ dst, int E,
                        const float* __restrict__ asrc, const float* __restrict__ adst,
                        const float* __restrict__ m2, float* __restrict__ den2) {
  int e = blockIdx.x * blockDim.x + threadIdx.x;
  int EE = E + N_NODE;
  if (e >= EE) return;
  int s, d;
  if (e < E) { s = src[e]; d = dst[e]; } else { s = e - E; d = s; }
  atomicAdd(&den2[d], __expf(lrelu(asrc[s] + adst[d]) - m2[d]));
}

__global__ void k_escat2(const int* __restrict__ src, const int* __restrict__ dst, int E,
                         const float* __restrict__ asrc, const float* __restrict__ adst,
                         const float* __restrict__ m2, const float* __restrict__ den2,
                         const float* __restrict__ y2p, float* __restrict__ agg2) {
  int e = blockIdx.x * blockDim.x + threadIdx.x;
  int EE = E + N_NODE;
  if (e >= EE) return;
  int s, d;
  if (e < E) { s = src[e]; d = dst[e]; } else { s = e - E; d = s; }
  float v = lrelu(asrc[s] + adst[d]);
  float alpha = __expf(v - m2[d]) / (den2[d] + 1e-16f);
  float4 y = ((const float4*)(y2p + s * NFEAT))[0];
  float* og = agg2 + d * NFEAT;
  atomicAdd(&og[0], alpha * y.x);
  atomicAdd(&og[1], alpha * y.y);
  atomicAdd(&og[2], alpha * y.z);
  atomicAdd(&og[3], alpha * y.w);
}

// ---------------- bias + log_softmax ----------------
__global__ void k_final(const float* __restrict__ agg2, const float* __restrict__ bias2,
                        float* __restrict__ out) {
  int n = blockIdx.x * blockDim.x + threadIdx.x;
  if (n >= N_NODE) return;
  float v[4];
  float mx = -3.0e38f;
  #pragma unroll
  for (int c = 0; c < 4; ++c) {
    v[c] = agg2[n * 4 + c] + bias2[c];
    mx = fmaxf(mx, v[c]);
  }
  float s = 0.0f;
  #pragma unroll
  for (int c = 0; c < 4; ++c) s += __expf(v[c] - mx);
  float l = __logf(s);
  #pragma unroll
  for (int c = 0; c < 4; ++c) out[n * 4 + c] = v[c] - mx - l;
}

extern "C" void kernel_launch(void* const* d_in, const int* in_sizes, int n_in,
                              void* d_out, int out_size, void* d_ws, size_t ws_size,
                              hipStream_t stream) {
  const float* x    = (const float*)d_in[0];
  const int*   eidx = (const int*)d_in[1];
  const float* W1   = (const float*)d_in[2];
  const float* as1  = (const float*)d_in[3];
  const float* ad1  = (const float*)d_in[4];
  const float* b1   = (const float*)d_in[5];
  const float* W2   = (const float*)d_in[6];
  const float* as2  = (const float*)d_in[7];
  const float* ad2  = (const float*)d_in[8];
  const float* b2   = (const float*)d_in[9];
  float* out = (float*)d_out;

  const int E = in_sizes[1] / 2;
  const int* src = eidx;
  const int* dst = eidx + E;
  const int EE = E + N_NODE;

  // Workspace carve-up (floats); all per-step arrays fit in the 192MB L2.
  float* p = (float*)d_ws;
  float* xp1   = p; p += (size_t)N_NODE * HID;
  float* asrc1 = p; p += (size_t)N_NODE * HEADS;
  float* adst1 = p; p += (size_t)N_NODE * HEADS;
  float* m1    = p; p += (size_t)N_NODE * HEADS;
  float* den1  = p; p += (size_t)N_NODE * HEADS;
  float* agg1  = p; p += (size_t)N_NODE * HID;
  float* h1    = p; p += (size_t)N_NODE * HID;
  float* y2p   = p; p += (size_t)N_NODE * NFEAT;
  float* asrc2 = p; p += (size_t)N_NODE;
  float* adst2 = p; p += (size_t)N_NODE;
  float* m2    = p; p += (size_t)N_NODE;
  float* den2  = p; p += (size_t)N_NODE;
  float* agg2  = p; p += (size_t)N_NODE * NFEAT;
  size_t need = (size_t)(p - (float*)d_ws) * sizeof(float);
  if (need > ws_size) return;  // deterministic no-op if scratch is too small

  const int B = 256;
  const int tiles = N_NODE / 16;                       // exact
  const int tblocks = (tiles + (B / 32) - 1) / (B / 32);
  const int gInit  = (N_NODE * HID + B - 1) / B;
  const int gA1    = (N_NODE * HEADS + B - 1) / B;
  const int gE     = (EE + B - 1) / B;
  const int gE8    = (EE * 8 + B - 1) / B;
  const int gN     = (N_NODE + B - 1) / B;

  for (int t = 0; t < SEQN; ++t) {
    const float* xt = x + (size_t)t * N_NODE * NFEAT;
    float* outt = out + (size_t)t * N_NODE * NFEAT;

    k_init  <<<gInit, B, 0, stream>>>(m1, den1, agg1, m2, den2, agg2);
    k_xform1<<<tblocks, B, 0, stream>>>(xt, W1, xp1);
    k_alpha1<<<gA1, B, 0, stream>>>(xp1, as1, ad1, asrc1, adst1);
    k_emax1 <<<gE, B, 0, stream>>>(src, dst, E, asrc1, adst1, m1);
    k_eden1 <<<gE, B, 0, stream>>>(src, dst, E, asrc1, adst1, m1, den1);
    k_escat1<<<gE8, B, 0, stream>>>(src, dst, E, asrc1, adst1, m1, den1, xp1, agg1);
    k_elu   <<<gInit, B, 0, stream>>>(agg1, b1, h1);
    k_xform2<<<tblocks, B, 0, stream>>>(h1, W2, y2p);
    k_alpha2<<<gN, B, 0, stream>>>(y2p, as2, ad2, asrc2, adst2);
    k_emax2 <<<gE, B, 0, stream>>>(src, dst, E, asrc2, adst2, m2);
    k_eden2 <<<gE, B, 0, stream>>>(src, dst, E, asrc2, adst2, m2, den2);
    k_escat2<<<gE, B, 0, stream>>>(src, dst, E, asrc2, adst2, m2, den2, y2p, agg2);
    k_final <<<gN, B, 0, stream>>>(agg2, b2, outt);
  }
}